// se1d_12670153523753
// MI455X (gfx1250) — compile-verified
//
#include <hip/hip_runtime.h>
#include <math.h>

typedef __attribute__((ext_vector_type(2))) float v2f;
typedef __attribute__((ext_vector_type(8))) float v8f;

#define HW   3136   // 56*56
#define CCH  512
#define NB   32     // batch
#define EPSV 1e-5f

// ---------------------------------------------------------------------------
// Kernel 1: per-(b,c) streaming reduction of x -> s = mean(x), q = mean(x^2)
// One block per (b,c) row (3136 contiguous floats, 16B-aligned).
// This is the only pass over the 205 MB tensor: bandwidth-bound, ~8.8us floor.
// ---------------------------------------------------------------------------
__global__ void se_reduce_kernel(const float* __restrict__ x,
                                 float* __restrict__ s_out,
                                 float* __restrict__ q_out) {
    const int bc = blockIdx.x;  // 0 .. 32*512-1
    const float4* p = (const float4*)(x + (size_t)bc * HW);
    float sum = 0.f, sq = 0.f;
    for (int i = threadIdx.x; i < HW / 4; i += blockDim.x) {
        float4 v = p[i];
        sum += v.x + v.y + v.z + v.w;
        sq  += v.x * v.x + v.y * v.y + v.z * v.z + v.w * v.w;
    }
    // wave32 reduction
    #pragma unroll
    for (int off = 16; off > 0; off >>= 1) {
        sum += __shfl_down(sum, off, 32);
        sq  += __shfl_down(sq,  off, 32);
    }
    __shared__ float ls[8], lq[8];
    const int lane = threadIdx.x & 31;
    const int wv   = threadIdx.x >> 5;
    if (lane == 0) { ls[wv] = sum; lq[wv] = sq; }
    __syncthreads();
    if (threadIdx.x == 0) {
        float ts = 0.f, tq = 0.f;
        #pragma unroll
        for (int i = 0; i < 8; ++i) { ts += ls[i]; tq += lq[i]; }
        const float inv = 1.0f / (float)HW;
        s_out[bc] = ts * inv;
        q_out[bc] = tq * inv;
    }
}

// ---------------------------------------------------------------------------
// Kernel 2: excitation via f32 WMMA (V_WMMA_F32_16X16X4_F32), single WG.
//   t1 = relu(s @ w1^T)            [32 x 32],  K = 512
//   e  = sigmoid(t1 @ w2^T)        [32 x 512], K = 32
//   my = s * e ;  vv = e^2 * q     (per-(b,c) spatial moments of y)
// 4 waves; phase 1: one 16x16 tile per wave; phase 2: 16 tiles per wave.
// f32 16x16x4 layouts: A lane<16 holds (row=l, K=0,1), lane>=16 (row=l, K=2,3);
// B symmetric over N; C/D: VGPR r = row 16*mt + r + 8*half, col 16*nt + l.
// ---------------------------------------------------------------------------
__global__ void se_excite_kernel(const float* __restrict__ s,
                                 const float* __restrict__ q,
                                 const float* __restrict__ w1,
                                 const float* __restrict__ w2,
                                 float* __restrict__ my,
                                 float* __restrict__ vv) {
    __shared__ float t1[32][33];  // padded: avoid bank conflicts on K-major reads
    const int lane = threadIdx.x & 31;
    const int wv   = threadIdx.x >> 5;   // 0..3
    const int half = lane >> 4;          // 0/1 -> K sub-pair
    const int l    = lane & 15;

    // ---- phase 1: t1 = relu(s @ w1^T) ----
    {
        const int mt = wv >> 1, nt = wv & 1;
        const int arow = 16 * mt + l;  // batch row
        const int brow = 16 * nt + l;  // output col n (row of w1)
        v8f acc = {};
        for (int k = 0; k < CCH; k += 4) {
            v2f a, b;
            a.x = s[arow * CCH + k + 2 * half + 0];
            a.y = s[arow * CCH + k + 2 * half + 1];
            b.x = w1[brow * CCH + k + 2 * half + 0];
            b.y = w1[brow * CCH + k + 2 * half + 1];
            acc = __builtin_amdgcn_wmma_f32_16x16x4_f32(
                false, a, false, b, (short)0, acc, false, false);
        }
        #pragma unroll
        for (int r = 0; r < 8; ++r) {
            float vr = acc[r];
            t1[16 * mt + r + 8 * half][16 * nt + l] = vr > 0.f ? vr : 0.f;
        }
    }
    __syncthreads();

    // ---- phase 2: e = sigmoid(t1 @ w2^T); emit my, vv ----
    for (int tile = wv; tile < 64; tile += 4) {
        const int mt = tile & 1;    // batch tile 0..1
        const int nt = tile >> 1;   // channel tile 0..31
        const int arow = 16 * mt + l;
        const int ncol = 16 * nt + l;  // global channel for this lane
        v8f acc = {};
        #pragma unroll
        for (int k = 0; k < 32; k += 4) {
            v2f a, b;
            a.x = t1[arow][k + 2 * half + 0];
            a.y = t1[arow][k + 2 * half + 1];
            b.x = w2[ncol * 32 + k + 2 * half + 0];
            b.y = w2[ncol * 32 + k + 2 * half + 1];
            acc = __builtin_amdgcn_wmma_f32_16x16x4_f32(
                false, a, false, b, (short)0, acc, false, false);
        }
        #pragma unroll
        for (int r = 0; r < 8; ++r) {
            const int b = 16 * mt + r + 8 * half;
            const int c = ncol;
            float e  = 1.0f / (1.0f + __expf(-acc[r]));
            float sv = s[b * CCH + c];
            float qv = q[b * CCH + c];
            my[b * CCH + c] = sv * e;
            vv[b * CCH + c] = e * e * qv;
        }
    }
}

// ---------------------------------------------------------------------------
// Kernel 3: BN stats over batch + channel-pair sum.
//   mu_c = mean_b my ; var_c = mean_b vv - mu^2 ; g_c = bn_w * rsqrt(var+eps)
//   out[b,k] = (my[b,k]-mu_k)g_k + (my[b,k+256]-mu_{k+256})g_{k+256} + bias_k+bias_{k+256}
// ---------------------------------------------------------------------------
__global__ void se_finalize_kernel(const float* __restrict__ my,
                                   const float* __restrict__ vv,
                                   const float* __restrict__ bw,
                                   const float* __restrict__ bb,
                                   float* __restrict__ out) {
    const int k = threadIdx.x;  // 0..255 (one block of 256)
    float s1 = 0.f, t1 = 0.f, s2 = 0.f, t2 = 0.f;
    #pragma unroll 4
    for (int b = 0; b < NB; ++b) {
        s1 += my[b * CCH + k];
        t1 += vv[b * CCH + k];
        s2 += my[b * CCH + k + 256];
        t2 += vv[b * CCH + k + 256];
    }
    const float inv = 1.0f / (float)NB;
    const float mu1 = s1 * inv, mu2 = s2 * inv;
    const float g1 = bw[k]       * rsqrtf(t1 * inv - mu1 * mu1 + EPSV);
    const float g2 = bw[k + 256] * rsqrtf(t2 * inv - mu2 * mu2 + EPSV);
    const float bias = bb[k] + bb[k + 256];
    for (int b = 0; b < NB; ++b) {
        out[b * 256 + k] = (my[b * CCH + k] - mu1) * g1
                         + (my[b * CCH + k + 256] - mu2) * g2 + bias;
    }
}

// ---------------------------------------------------------------------------
extern "C" void kernel_launch(void* const* d_in, const int* in_sizes, int n_in,
                              void* d_out, int out_size, void* d_ws, size_t ws_size,
                              hipStream_t stream) {
    const float* x  = (const float*)d_in[0];  // [32,512,56,56]
    const float* w1 = (const float*)d_in[1];  // [32,512]
    const float* w2 = (const float*)d_in[2];  // [512,32]
    const float* bw = (const float*)d_in[3];  // [512]
    const float* bb = (const float*)d_in[4];  // [512]
    float* out = (float*)d_out;               // [32,256]

    float* ws = (float*)d_ws;                 // needs 4*16384 floats = 256 KB
    float* s  = ws;                           // [32,512] mean(x)
    float* q  = ws + 16384;                   // [32,512] mean(x^2)
    float* my = ws + 32768;                   // [32,512] mean(y)
    float* vv = ws + 49152;                   // [32,512] mean(y^2)

    se_reduce_kernel<<<NB * CCH, 256, 0, stream>>>(x, s, q);
    se_excite_kernel<<<1, 128, 0, stream>>>(s, q, w1, w2, my, vv);
    se_finalize_kernel<<<1, 256, 0, stream>>>(my, vv, bw, bb, out);
}